// SpatialEmbLoss_2D_87187836109473
// MI455X (gfx1250) — compile-verified
//
#include <hip/hip_runtime.h>
#include <hip/hip_bf16.h>
#include <math.h>

// ---------------------------------------------------------------------------
// SpatialEmbLoss (B=4, C=5, H=W=1024, MAX_ID=16) for MI455X / gfx1250
// Multi-kernel pipeline:
//   Z: zero workspace
//   A: per-instance stats (count, sum x, sum y, sum sigma0/1, cm stats) + seed_bg
//   B: per-instance params (center, s, exp(10 s))
//   C: per-(b,id) 4096-bin error histogram (+ var & seed_fg accumulators)
//   D: Lovasz via binned telescoping Jaccard; prefix scans done with
//      V_WMMA_F32_16X16X4_F32 (D = L * X, L = lower-triangular ones)
//   E: final reduction to a scalar
// ---------------------------------------------------------------------------

#define B_   4
#define HW_  (1024 * 1024)
#define NB_  4096
#define NSIG 2

// workspace layout (in floats)
#define OFF_STATS   0                               // B*16*8 : cnt,sx,sy,ss0,ss1,cmc,cmx,cmy
#define OFF_PARAMS  (OFF_STATS  + B_ * 16 * 8)      // B*16*8 : cnt,cx,cy,s0,s1,se0,se1,pad
#define OFF_HISTC   (OFF_PARAMS + B_ * 16 * 8)      // B*16*NB : count per bin
#define OFF_HISTG   (OFF_HISTC  + B_ * 16 * NB_)    // B*16*NB : gt count per bin
#define OFF_HISTE   (OFF_HISTG  + B_ * 16 * NB_)    // B*16*NB : sum(err) per bin
#define OFF_ACC     (OFF_HISTE  + B_ * 16 * NB_)    // B*16*3  : var_raw, seed_fg, inst
#define OFF_SEEDBG  (OFF_ACC    + B_ * 16 * 3)      // B_
#define WS_FLOATS   (OFF_SEEDBG + B_)

typedef float v2f __attribute__((ext_vector_type(2)));
typedef float v8f __attribute__((ext_vector_type(8)));

__device__ __forceinline__ float sigmoidf_(float x) { return 1.0f / (1.0f + __expf(-x)); }

// ---------------------------------------------------------------------------
// Z: zero the workspace (graph-replay safe: re-zeroed every launch)
// ---------------------------------------------------------------------------
__global__ void zero_kernel(float* ws, int n) {
  int i = blockIdx.x * blockDim.x + threadIdx.x;
  if (i < n) ws[i] = 0.0f;
}

// ---------------------------------------------------------------------------
// A: per-instance stats + seed background loss
// grid (256, B), block 256; each block owns 4096 consecutive pixels
// ---------------------------------------------------------------------------
__global__ __launch_bounds__(256) void stats_kernel(
    const float* __restrict__ pred, const int* __restrict__ inst,
    const int* __restrict__ lab, const unsigned char* __restrict__ cent,
    float* __restrict__ ws) {
  __shared__ float sacc[16 * 8];
  __shared__ float sbg;
  const int tid = threadIdx.x;
  const int bi  = blockIdx.y;
  if (tid < 128) sacc[tid] = 0.0f;
  if (tid == 0) sbg = 0.0f;
  __syncthreads();

  const float* p2 = pred + ((size_t)bi * 5 + 2) * HW_;
  const float* p3 = pred + ((size_t)bi * 5 + 3) * HW_;
  const float* p4 = pred + ((size_t)bi * 5 + 4) * HW_;
  const int*   ip = inst + (size_t)bi * HW_;
  const int*   lp = lab  + (size_t)bi * HW_;
  const unsigned char* cp = cent + (size_t)bi * HW_;

  float bgAcc = 0.0f;
  const int base = blockIdx.x * 4096;
#pragma unroll 4
  for (int k = 0; k < 16; ++k) {
    const int p = base + k * 256 + tid;
    const int x = p & 1023, y = p >> 10;
    const float xm = (float)x * (1.0f / 1023.0f);
    const float ym = (float)y * (1.0f / 1023.0f);
    const int iid = ip[p];
    const float seed = sigmoidf_(p4[p]);
    if (lp[p] == 0) bgAcc += seed * seed;
    if (iid >= 1) {
      const int b8 = (iid - 1) * 8;
      atomicAdd(&sacc[b8 + 0], 1.0f);
      atomicAdd(&sacc[b8 + 1], xm);
      atomicAdd(&sacc[b8 + 2], ym);
      atomicAdd(&sacc[b8 + 3], p2[p]);
      atomicAdd(&sacc[b8 + 4], p3[p]);
      if (cp[p]) {  // center-image stats (all-false in harness, kept for fidelity)
        atomicAdd(&sacc[b8 + 5], 1.0f);
        atomicAdd(&sacc[b8 + 6], xm);
        atomicAdd(&sacc[b8 + 7], ym);
      }
    }
  }
  atomicAdd(&sbg, bgAcc);
  __syncthreads();
  if (tid < 128) {
    const float v = sacc[tid];
    if (v != 0.0f) atomicAdd(&ws[OFF_STATS + bi * 128 + tid], v);
  }
  if (tid == 0) atomicAdd(&ws[OFF_SEEDBG + bi], sbg);
}

// ---------------------------------------------------------------------------
// B: per-instance parameters
// ---------------------------------------------------------------------------
__global__ void params_kernel(float* __restrict__ ws) {
  const int t = threadIdx.x;
  if (t >= B_ * 16) return;
  const float* st = ws + OFF_STATS + t * 8;
  const float cnt = st[0];
  const float safe = fmaxf(cnt, 1.0f);
  const float cmc = st[5];
  const float cx = (cmc == 1.0f) ? st[6] : st[1] / safe;
  const float cy = (cmc == 1.0f) ? st[7] : st[2] / safe;
  const float s0 = st[3] / safe;
  const float s1 = st[4] / safe;
  float* pr = ws + OFF_PARAMS + t * 8;
  pr[0] = cnt; pr[1] = cx; pr[2] = cy;
  pr[3] = s0;  pr[4] = s1;
  pr[5] = __expf(10.0f * s0);
  pr[6] = __expf(10.0f * s1);
  pr[7] = 0.0f;
}

// ---------------------------------------------------------------------------
// C: per-(b,id) binned error histogram + var + seed_fg
// grid (32, 16, B), block 256. LDS histogram = 48 KB (of 320 KB/WGP).
// Per-image working set (24 MB) fits in 192 MB L2 -> 15 of 16 id-passes
// replay from L2, HBM traffic ~96 MB total (~4 us at 23.3 TB/s).
// ---------------------------------------------------------------------------
__global__ __launch_bounds__(256) void hist_kernel(
    const float* __restrict__ pred, const int* __restrict__ inst,
    float* __restrict__ ws) {
  const int bi  = blockIdx.z;
  const int idm = blockIdx.y;
  const int id  = idm + 1;
  const float* prm = ws + OFF_PARAMS + (bi * 16 + idm) * 8;
  const float cnt = prm[0];
  if (cnt < 0.5f) return;  // instance absent: all its terms are zero
  const float cx = prm[1], cy = prm[2];
  const float s0 = prm[3], s1 = prm[4];
  const float se0 = prm[5], se1 = prm[6];

  __shared__ float hC[NB_];
  __shared__ float hG[NB_];
  __shared__ float hE[NB_];
  __shared__ float redV[256];
  __shared__ float redS[256];
  const int tid = threadIdx.x;
  for (int t = tid; t < NB_; t += 256) { hC[t] = 0.0f; hG[t] = 0.0f; hE[t] = 0.0f; }
  __syncthreads();

  const float* p0 = pred + ((size_t)bi * 5 + 0) * HW_;
  const float* p1 = pred + ((size_t)bi * 5 + 1) * HW_;
  const float* p2 = pred + ((size_t)bi * 5 + 2) * HW_;
  const float* p3 = pred + ((size_t)bi * 5 + 3) * HW_;
  const float* p4 = pred + ((size_t)bi * 5 + 4) * HW_;
  const int*   ip = inst + (size_t)bi * HW_;

  float varAcc = 0.0f, sfAcc = 0.0f;
  const int start = blockIdx.x * (HW_ / 32);
  const int end   = start + (HW_ / 32);
  for (int p = start + tid; p < end; p += 256) {
    // hint the streaming pattern to GL2 (lowers to global_prefetch_b8)
    __builtin_prefetch(&p0[p + 2048], 0, 1);
    __builtin_prefetch(&ip[p + 2048], 0, 1);
    const int x = p & 1023, y = p >> 10;
    const float xm = (float)x * (1.0f / 1023.0f);
    const float ym = (float)y * (1.0f / 1023.0f);
    const float ex = tanhf(p0[p]) + xm;
    const float ey = tanhf(p1[p]) + ym;
    const float dx = ex - cx, dy = ey - cy;
    const float dist = __expf(-(dx * dx * se0 + dy * dy * se1));
    const float logit = 2.0f * dist - 1.0f;
    const bool  m = (ip[p] == id);
    const float err = 1.0f - logit * (m ? 1.0f : -1.0f);  // in [0,2]
    int bin = (int)(err * (float)(NB_ / 2));
    bin = bin < 0 ? 0 : (bin > NB_ - 1 ? NB_ - 1 : bin);
    atomicAdd(&hC[bin], 1.0f);
    atomicAdd(&hE[bin], err);
    if (m) {
      atomicAdd(&hG[bin], 1.0f);
      const float d2c = p2[p] - s0, d3c = p3[p] - s1;
      varAcc += d2c * d2c + d3c * d3c;
      const float seed = sigmoidf_(p4[p]);
      const float df = seed - dist;
      sfAcc += df * df;
    }
  }
  redV[tid] = varAcc; redS[tid] = sfAcc;
  __syncthreads();
  for (int s = 128; s > 0; s >>= 1) {
    if (tid < s) { redV[tid] += redV[tid + s]; redS[tid] += redS[tid + s]; }
    __syncthreads();
  }
  float* acc = ws + OFF_ACC + (bi * 16 + idm) * 3;
  if (tid == 0) { atomicAdd(&acc[0], redV[0]); atomicAdd(&acc[1], redS[0]); }
  const int gb = (bi * 16 + idm) * NB_;
  for (int t = tid; t < NB_; t += 256) {
    if (hC[t] != 0.0f) {
      atomicAdd(&ws[OFF_HISTC + gb + t], hC[t]);
      atomicAdd(&ws[OFF_HISTE + gb + t], hE[t]);
      if (hG[t] != 0.0f) atomicAdd(&ws[OFF_HISTG + gb + t], hG[t]);
    }
  }
}

// ---------------------------------------------------------------------------
// D: binned Lovasz hinge. One wave (32 lanes) per (b,id).
// Inclusive prefix sums of per-bin counts (descending-error order) are
// computed 256 bins at a time as D = L * X with V_WMMA_F32_16X16X4_F32:
//   L = 16x16 lower-triangular ones (A operand, 4 chained K=4 slices)
//   X = 16 segments of 16 bins, column-major (B operand)
// f32 WMMA keeps the integer-valued counts (<= 2^20) exact.
// VGPR layouts per ISA 7.12.2:
//   A 16x4 f32: lanes 0-15 rows M=0..15 w/ K={4c,4c+1}; lanes 16-31 K={4c+2,4c+3}
//   B 4x16 f32: lane supplies N=lane&15 for the same K pair
//   C/D 16x16:  vgpr v, lane l -> M = v + 8*(l>>4), N = l&15
// ---------------------------------------------------------------------------
__global__ __launch_bounds__(32) void lovasz_kernel(float* __restrict__ ws) {
  const int gid = blockIdx.x;          // b*16 + idm
  const float G = ws[OFF_PARAMS + gid * 8 + 0];
  if (G < 0.5f) return;                // absent instance: acc already zero

  __shared__ float cA[256], cG[256], sE[256];
  __shared__ float oC[256], oG[256];
  __shared__ float segC[17], segG[17];

  const int lane = threadIdx.x;
  const int nn = lane & 15;            // N column / A row
  const int hh = lane >> 4;            // lane-half selects K pair
  const float* hCg = ws + OFF_HISTC + gid * NB_;
  const float* hGg = ws + OFF_HISTG + gid * NB_;
  const float* hEg = ws + OFF_HISTE + gid * NB_;

  float carryC = 0.0f, carryG = 0.0f, loss = 0.0f;

  for (int chunk = 0; chunk < 16; ++chunk) {
    for (int t = lane; t < 256; t += 32) {
      const int bin = NB_ - 1 - (chunk * 256 + t);   // descending error order
      cA[t] = hCg[bin]; cG[t] = hGg[bin]; sE[t] = hEg[bin];
    }
    __syncthreads();

    v8f dC = {}; v8f dG = {};
#pragma unroll
    for (int c = 0; c < 4; ++c) {
      const int k0 = 4 * c + 2 * hh;                 // global K of this lane's pair
      v2f a;                                         // L[nn][k0], L[nn][k0+1]
      a.x = (k0     <= nn) ? 1.0f : 0.0f;
      a.y = (k0 + 1 <= nn) ? 1.0f : 0.0f;
      v2f bC; bC.x = cA[nn * 16 + k0]; bC.y = cA[nn * 16 + k0 + 1];
      v2f bG; bG.x = cG[nn * 16 + k0]; bG.y = cG[nn * 16 + k0 + 1];
      dC = __builtin_amdgcn_wmma_f32_16x16x4_f32(false, a, false, bC, (short)0, dC, false, false);
      dG = __builtin_amdgcn_wmma_f32_16x16x4_f32(false, a, false, bG, (short)0, dG, false, false);
    }
#pragma unroll
    for (int v = 0; v < 8; ++v) {                    // D -> LDS (column-major)
      const int m = v + 8 * hh;
      oC[nn * 16 + m] = dC[v];
      oG[nn * 16 + m] = dG[v];
    }
    __syncthreads();

    if (lane == 0) {                                 // cross-segment carries
      float offc = carryC, offg = carryG;
      for (int n = 0; n < 16; ++n) {
        segC[n] = offc; offc += oC[n * 16 + 15];
        segG[n] = offg; offg += oG[n * 16 + 15];
      }
      segC[16] = offc; segG[16] = offg;
    }
    __syncthreads();
    carryC = segC[16]; carryG = segG[16];

    for (int t = lane; t < 256; t += 32) {
      const float cnt = cA[t];
      if (cnt > 0.0f) {
        const float cum  = oC[t] + segC[t >> 4];     // inclusive rank
        const float cumg = oG[t] + segG[t >> 4];     // inclusive gt count
        const float rp = cum - cnt, cgp = cumg - cG[t];
        const float jacA = 1.0f - (G - cumg) / (G + cum - cumg);
        const float jacB = 1.0f - (G - cgp)  / (G + rp  - cgp);
        const float me = sE[t] / cnt;                // errors >= 0 -> relu no-op
        loss += fmaxf(me, 0.0f) * (jacA - jacB);
      }
    }
    __syncthreads();
  }

#pragma unroll
  for (int off = 16; off > 0; off >>= 1) loss += __shfl_down(loss, off, 32);
  if (lane == 0) ws[OFF_ACC + gid * 3 + 2] = loss;
}

// ---------------------------------------------------------------------------
// E: final combine
// ---------------------------------------------------------------------------
__global__ void final_kernel(const float* __restrict__ ws, float* __restrict__ out) {
  __shared__ float sI[B_], sV[B_], sS[B_], sP[B_], lb[B_];
  const int t = threadIdx.x;
  if (t < B_) { sI[t] = 0.0f; sV[t] = 0.0f; sS[t] = 0.0f; sP[t] = 0.0f; }
  __syncthreads();
  if (t < B_ * 16) {
    const int bi = t >> 4;
    const float cnt = ws[OFF_PARAMS + t * 8 + 0];
    if (cnt > 0.5f) {
      const float safe = fmaxf(cnt, 1.0f);
      atomicAdd(&sI[bi], ws[OFF_ACC + t * 3 + 2]);
      atomicAdd(&sV[bi], ws[OFF_ACC + t * 3 + 0] / (safe * (float)NSIG));
      atomicAdd(&sS[bi], ws[OFF_ACC + t * 3 + 1]);
      atomicAdd(&sP[bi], 1.0f);
    }
  }
  __syncthreads();
  if (t < B_) {
    const float denom = fmaxf(sP[t], 1.0f);
    const float seed_loss = (ws[OFF_SEEDBG + t] + sS[t]) * (1.0f / (float)HW_);
    lb[t] = 1.0f * (sI[t] / denom) + 10.0f * (sV[t] / denom) + 1.0f * seed_loss;
  }
  __syncthreads();
  if (t == 0) out[0] = (lb[0] + lb[1] + lb[2] + lb[3]) * (1.0f / (float)B_);
}

// ---------------------------------------------------------------------------
extern "C" void kernel_launch(void* const* d_in, const int* in_sizes, int n_in,
                              void* d_out, int out_size, void* d_ws, size_t ws_size,
                              hipStream_t stream) {
  const float* pred = (const float*)d_in[0];
  const int*   inst = (const int*)d_in[1];
  const int*   lab  = (const int*)d_in[2];
  const unsigned char* cent = (const unsigned char*)d_in[3];
  float* ws  = (float*)d_ws;
  float* out = (float*)d_out;

  zero_kernel<<<(WS_FLOATS + 255) / 256, 256, 0, stream>>>(ws, WS_FLOATS);
  stats_kernel<<<dim3(256, B_), 256, 0, stream>>>(pred, inst, lab, cent, ws);
  params_kernel<<<1, 64, 0, stream>>>(ws);
  hist_kernel<<<dim3(32, 16, B_), 256, 0, stream>>>(pred, inst, ws);
  lovasz_kernel<<<B_ * 16, 32, 0, stream>>>(ws);
  final_kernel<<<1, 64, 0, stream>>>(ws, out);
}